// Shift_gcn_local_23261542875779
// MI455X (gfx1250) — compile-verified
//
#include <hip/hip_runtime.h>

#define Nn   64
#define Cc   64
#define Tt   300
#define Vv   25
#define COo  64
#define NT   (Nn * Tt)        // 19200 rows for BN
#define Ff   (Vv * Cc)        // 1600 features
#define Mrows (NT * Vv)       // 480000 GEMM rows
#define EPSc 1e-5f

typedef __attribute__((ext_vector_type(2))) float v2f;
typedef __attribute__((ext_vector_type(8))) float v8f;

// ---------------------------------------------------------------------------
// Kernel 0: precompute gather BYTE offsets into x's (N,C,T,V) layout and the
// (tanh(mask)+1) scale, both indexed by the post-shift feature j = v*C + c.
//   shift_in[j] = v'*C + c'  ->  byte offset within an (n,t) slab:
//   (c'*T*V + v') * 4
// ---------------------------------------------------------------------------
__global__ void k_setup(const int* __restrict__ shift_in,
                        const float* __restrict__ feat_mask,
                        int* __restrict__ soff, float* __restrict__ mval) {
    int j = blockIdx.x * 256 + threadIdx.x;
    if (j < Ff) {
        int f  = shift_in[j];
        int vp = f / Cc;
        int cp = f % Cc;
        soff[j] = (cp * (Tt * Vv) + vp) * 4;
        mval[j] = tanhf(feat_mask[j]) + 1.0f;
    }
}

// ---------------------------------------------------------------------------
// Kernel 1: fused gather + mask + GEMM (M=480000, K=64, N=64) via
// V_WMMA_F32_16X16X4_F32.  8 waves/block, each wave: 16-row tile x 4 N-tiles.
// lin_w staged in LDS PRE-PAIRED as float2 so every B fragment is a single
// ds_load_b64 with an immediate offset.  All x gathers use a uniform SADDR
// base + 32-bit per-lane byte offset (x is 123 MB < 2^31), so each scattered
// load costs exactly one v_add_nc_u32 of addressing.
// A layout (16x4 f32): lanes 0-15 -> M=lane, K pair (4s,4s+1);
//                      lanes 16-31 -> M=lane-16, K pair (4s+2,4s+3).
// D layout: VGPR i -> M = i (lanes 0-15) / 8+i (lanes 16-31), N = lane&15.
// ---------------------------------------------------------------------------
__global__ __launch_bounds__(256) void k_gemm(const float* __restrict__ x,
                                              const float* __restrict__ lin_w,
                                              const float* __restrict__ lin_b,
                                              const int* __restrict__ soff,
                                              const float* __restrict__ mval,
                                              float* __restrict__ y) {
    __shared__ v2f   lw2[32 * COo];  // pair p = c0/2 in [0,32), d in [0,64) : 16 KB
    __shared__ int   ls[Ff];         // 6.4 KB (byte offsets)
    __shared__ float lm[Ff];         // 6.4 KB

    int tid = threadIdx.x;
    for (int i = tid; i < 32 * COo; i += 256) {
        int p = i >> 6;
        int d = i & 63;
        v2f w;
        w.x = lin_w[(2 * p) * COo + d];
        w.y = lin_w[(2 * p + 1) * COo + d];
        lw2[i] = w;
    }
    for (int i = tid; i < Ff; i += 256) { ls[i] = soff[i]; lm[i] = mval[i]; }
    __syncthreads();

    int wave  = tid >> 5;
    int lane  = tid & 31;
    int lhalf = lane >> 4;    // which K-half this lane carries
    int l16   = lane & 15;
    int rbase = (blockIdx.x * 8 + wave) * 16;

    // This lane's A-matrix row (M = rbase + l16)
    int row = rbase + l16;
    int nt  = row / Vv;
    int v   = row % Vv;
    int n   = nt / Tt;
    int t   = nt % Tt;
    // per-lane 32-bit byte offset of this (n,t) slab within x (max ~123 MB)
    unsigned xb = (unsigned)((n * (Cc * Tt * Vv) + t * Vv) * 4);
    const char* xc = (const char*)x;
    const int*   lsv = ls + v * Cc;
    const float* lmv = lm + v * Cc;

    v8f acc0 = {}, acc1 = {}, acc2 = {}, acc3 = {};

#pragma unroll
    for (int s = 0; s < 16; ++s) {
        int c0 = 4 * s + 2 * lhalf;
        unsigned o0 = xb + (unsigned)lsv[c0];
        unsigned o1 = xb + (unsigned)lsv[c0 + 1];
        v2f a;
        a.x = *(const float*)(xc + (size_t)o0) * lmv[c0];
        a.y = *(const float*)(xc + (size_t)o1) * lmv[c0 + 1];

        // B fragments: pair index p = c0/2 = 2s + lhalf
        const v2f* wp = lw2 + (2 * s + lhalf) * COo + l16;
        v2f b0 = wp[0];
        v2f b1 = wp[16];
        v2f b2 = wp[32];
        v2f b3 = wp[48];

        acc0 = __builtin_amdgcn_wmma_f32_16x16x4_f32(false, a, false, b0, (short)0, acc0, false, false);
        acc1 = __builtin_amdgcn_wmma_f32_16x16x4_f32(false, a, false, b1, (short)0, acc1, false, false);
        acc2 = __builtin_amdgcn_wmma_f32_16x16x4_f32(false, a, false, b2, (short)0, acc2, false, false);
        acc3 = __builtin_amdgcn_wmma_f32_16x16x4_f32(false, a, false, b3, (short)0, acc3, false, false);
    }

    // Store D tiles (+ bias) to y workspace: y[(nt*25+v)*64 + d] == y[nt][v*64+d]
    int   orow  = rbase + 8 * lhalf;
    int   ocol  = l16;
    float bias0 = lin_b[ocol];
    float bias1 = lin_b[ocol + 16];
    float bias2 = lin_b[ocol + 32];
    float bias3 = lin_b[ocol + 48];
#pragma unroll
    for (int i = 0; i < 8; ++i) {
        long o = (long)(orow + i) * COo + ocol;
        y[o]      = acc0[i] + bias0;
        y[o + 16] = acc1[i] + bias1;
        y[o + 32] = acc2[i] + bias2;
        y[o + 48] = acc3[i] + bias3;
    }
}

// ---------------------------------------------------------------------------
// Kernel 2: per-column sum / sum-of-squares over the 19200 BN rows.
// 32 adjacent columns per block (lanes -> columns, coalesced 128B reads),
// 8 waves stride over rows; cross-wave reduce in LDS.  No atomics.
// ---------------------------------------------------------------------------
__global__ __launch_bounds__(256) void k_stats(const float* __restrict__ y,
                                               float* __restrict__ csum,
                                               float* __restrict__ csq) {
    __shared__ float ss[8][32];
    __shared__ float sq[8][32];
    int lane = threadIdx.x & 31;
    int w    = threadIdx.x >> 5;
    int f    = blockIdx.x * 32 + lane;

    float s = 0.f, q = 0.f;
    for (int nt = w; nt < NT; nt += 8) {
        float val = y[(long)nt * Ff + f];
        s += val;
        q += val * val;
    }
    ss[w][lane] = s;
    sq[w][lane] = q;
    __syncthreads();
    if (w == 0) {
#pragma unroll
        for (int i = 1; i < 8; ++i) {
            s += ss[i][lane];
            q += sq[i][lane];
        }
        csum[f] = s;
        csq[f]  = q;
    }
}

// ---------------------------------------------------------------------------
// Kernel 3: fold shift_out + BN into per-output-column affine (a,b):
//   z_norm[k] = a[k]*y[shift_out[k]] + b[k]
// Also pre-scale the gather index to a byte offset for the epilogue.
// ---------------------------------------------------------------------------
__global__ void k_finalize(const int* __restrict__ shift_out,
                           const float* __restrict__ gamma,
                           const float* __restrict__ beta,
                           const float* __restrict__ csum,
                           const float* __restrict__ csq,
                           float* __restrict__ ak, float* __restrict__ bk,
                           int* __restrict__ sobyte) {
    int k = blockIdx.x * 256 + threadIdx.x;
    if (k < Ff) {
        int src = shift_out[k];
        float m   = csum[src] * (1.0f / NT);
        float var = csq[src] * (1.0f / NT) - m * m;
        float a   = gamma[k] * rsqrtf(var + EPSc);
        ak[k]     = a;
        bk[k]     = beta[k] - m * a;
        sobyte[k] = src * 4;
    }
}

// ---------------------------------------------------------------------------
// Kernel 4: gather + affine + residual + ReLU, coalesced over out's
// (N,CO,T,V) layout (which matches x exactly).  y gather uses uniform SADDR
// base + 32-bit per-lane byte offset (y is 123 MB < 2^31).
// ---------------------------------------------------------------------------
__global__ __launch_bounds__(256) void k_out(const float* __restrict__ x,
                                             const float* __restrict__ y,
                                             const int* __restrict__ sobyte,
                                             const float* __restrict__ ak,
                                             const float* __restrict__ bk,
                                             float* __restrict__ out) {
    long idx = (long)blockIdx.x * 256 + threadIdx.x;  // over N*CO*T*V
    int v   = (int)(idx % Vv);
    long r  = idx / Vv;
    int t   = (int)(r % Tt);
    long r2 = r / Tt;
    int co  = (int)(r2 % COo);
    int n   = (int)(r2 / COo);

    int k  = v * COo + co;
    int nt = n * Tt + t;
    unsigned yoff = (unsigned)(nt * Ff * 4) + (unsigned)sobyte[k];
    float yv  = *(const float*)((const char*)y + (size_t)yoff);
    float val = ak[k] * yv + bk[k] + x[idx];
    out[idx] = val > 0.f ? val : 0.f;
}

// ---------------------------------------------------------------------------
extern "C" void kernel_launch(void* const* d_in, const int* in_sizes, int n_in,
                              void* d_out, int out_size, void* d_ws, size_t ws_size,
                              hipStream_t stream) {
    const float* x         = (const float*)d_in[0];
    const float* lin_w     = (const float*)d_in[1];
    const float* lin_b     = (const float*)d_in[2];
    const float* feat_mask = (const float*)d_in[3];
    const float* bn_gamma  = (const float*)d_in[4];
    const float* bn_beta   = (const float*)d_in[5];
    const int*   shift_in  = (const int*)d_in[6];
    const int*   shift_out = (const int*)d_in[7];
    float* out = (float*)d_out;

    // workspace carve-up
    float* y_ws = (float*)d_ws;                                   // 480000*64 f32
    char*  p    = (char*)d_ws + (size_t)Mrows * COo * sizeof(float);
    int*   soff   = (int*)p;             p += Ff * sizeof(int);
    float* mval   = (float*)p;           p += Ff * sizeof(float);
    float* csum   = (float*)p;           p += Ff * sizeof(float);
    float* csq    = (float*)p;           p += Ff * sizeof(float);
    float* ak     = (float*)p;           p += Ff * sizeof(float);
    float* bk     = (float*)p;           p += Ff * sizeof(float);
    int*   sobyte = (int*)p;

    k_setup<<<(Ff + 255) / 256, 256, 0, stream>>>(shift_in, feat_mask, soff, mval);
    k_gemm<<<Mrows / 128, 256, 0, stream>>>(x, lin_w, lin_b, soff, mval, y_ws);
    k_stats<<<Ff / 32, 256, 0, stream>>>(y_ws, csum, csq);
    k_finalize<<<(Ff + 255) / 256, 256, 0, stream>>>(shift_out, bn_gamma, bn_beta,
                                                     csum, csq, ak, bk, sobyte);
    k_out<<<(int)(((long)Nn * COo * Tt * Vv) / 256), 256, 0, stream>>>(
        x, y_ws, sobyte, ak, bk, out);
}